// RspaA1_65695819759964
// MI455X (gfx1250) — compile-verified
//
#include <hip/hip_runtime.h>

typedef __attribute__((ext_vector_type(2))) float v2f;
typedef __attribute__((ext_vector_type(4))) float v4f;
typedef __attribute__((ext_vector_type(8))) float v8f;
typedef __attribute__((ext_vector_type(4))) int   v4i;

#define HW     4096
#define CDIM   64
#define KTILE  16
#define QTILE  16
#define WAVES  8
#define LOG2E  1.4426950408889634f

#if __has_builtin(__builtin_amdgcn_global_load_async_to_lds_b128)
#define HAS_ASYNC_LDS 1
#else
#define HAS_ASYNC_LDS 0
#endif

#if HAS_ASYNC_LDS
// Issue a 16B/lane async global->LDS copy (ASYNCcnt-tracked, no VGPR round trip).
__device__ __forceinline__ void async_fill_b128(const float* gsrc, float* ldst) {
    __builtin_amdgcn_global_load_async_to_lds_b128(
        (__attribute__((address_space(1))) v4i*)(uintptr_t)gsrc,
        (__attribute__((address_space(3))) v4i*)(uintptr_t)ldst,
        0, 0);
}
#endif

// DPP16 ROW_XMASK: lane ^= mask within each 16-lane row (gfx10+ DPP control).
template <int CTRL>
__device__ __forceinline__ float dpp_f32(float x) {
    return __int_as_float(
        __builtin_amdgcn_update_dpp(0, __float_as_int(x), CTRL, 0xf, 0xf, true));
}
#define ROW_XMASK(m) (0x160 | (m))

__device__ __forceinline__ float rowmax16(float x) {
    x = fmaxf(x, dpp_f32<ROW_XMASK(1)>(x));
    x = fmaxf(x, dpp_f32<ROW_XMASK(2)>(x));
    x = fmaxf(x, dpp_f32<ROW_XMASK(4)>(x));
    x = fmaxf(x, dpp_f32<ROW_XMASK(8)>(x));
    return x;
}
__device__ __forceinline__ float rowsum16(float x) {
    x += dpp_f32<ROW_XMASK(1)>(x);
    x += dpp_f32<ROW_XMASK(2)>(x);
    x += dpp_f32<ROW_XMASK(4)>(x);
    x += dpp_f32<ROW_XMASK(8)>(x);
    return x;
}

// Fused self-attention: per batch, S = X^T X (C=64 inner), row softmax, out = P X^T.
// fp32 end-to-end via V_WMMA_F32_16X16X4_F32; flash-style (no score materialization).
__global__ __launch_bounds__(256, 1) void RspaA1_attn_kernel(const float* __restrict__ X,
                                                             float* __restrict__ Out) {
    // Double-buffered K/V tile, row padded to 20 dwords:
    //  - 16B-aligned rows for b128 fill
    //  - conflict-free for lane-varies-m (score B) and lane-varies-c (PV B) reads
    __shared__ float ktile[2][CDIM][20];
    // Per-wave P tile for D-layout -> A-layout transpose (padded to 17).
    __shared__ float ptile[WAVES][16][17];

    const int n    = blockIdx.y;
    const int lane = threadIdx.x & 31;
    const int wave = threadIdx.x >> 5;
    const int ml   = lane & 15;     // column index within 16-lane half
    const int hi   = lane >> 4;     // which half
    const int coff = hi * 2;        // K-offset selector for WMMA A/B fragments

    const float* Xn = X   + (size_t)n * CDIM * HW;
    float*       On = Out + (size_t)n * CDIM * HW;

    const int qbase = blockIdx.x * (WAVES * QTILE) + wave * QTILE;

    // ---- Load Q fragments once (loop-invariant). A[q,k]: lane = q + 16*(k>=2), vgpr=k&1.
    v2f qa[16];
    {
        const int q = qbase + ml;
        #pragma unroll
        for (int cb = 0; cb < 16; ++cb) {
            const int c0 = cb * 4 + coff;
            qa[cb].x = Xn[(size_t)(c0    ) * HW + q];
            qa[cb].y = Xn[(size_t)(c0 + 1) * HW + q];
        }
    }

    // Online-softmax state (in exp2 domain): slot v <-> row (v + 8*hi).
    float mrow[8], lrow[8];
    v8f acc[4];
    const v8f vzero = {0.f, 0.f, 0.f, 0.f, 0.f, 0.f, 0.f, 0.f};
    #pragma unroll
    for (int v = 0; v < 8; ++v) { mrow[v] = -3.0e38f; lrow[v] = 0.0f; }
    #pragma unroll
    for (int ct = 0; ct < 4; ++ct) acc[ct] = vzero;

    // Cooperative tile-fill mapping: 256 threads x one b128 = 64x16 fp32 tile.
    const int tc  = threadIdx.x >> 2;        // c row 0..63
    const int tm4 = (threadIdx.x & 3) * 4;   // m offset 0,4,8,12

    // ---- Preload tile 0.
#if HAS_ASYNC_LDS
    async_fill_b128(Xn + (size_t)tc * HW + tm4, &ktile[0][tc][tm4]);
    asm volatile("s_wait_asynccnt 0x0" ::: "memory");
#else
    *(v4f*)&ktile[0][tc][tm4] = *(const v4f*)(Xn + (size_t)tc * HW + tm4);
#endif
    __syncthreads();

    for (int mb = 0; mb < HW; mb += KTILE) {
        const int  cur = (mb >> 4) & 1;
        const bool pre = (mb + KTILE) < HW;

        // ---- Stage next K/V tile into the other buffer (consumers of that buffer
        // all completed before the previous barrier).
#if HAS_ASYNC_LDS
        if (pre) async_fill_b128(Xn + (size_t)tc * HW + (mb + KTILE) + tm4,
                                 &ktile[cur ^ 1][tc][tm4]);
#else
        v4f nv;
        if (pre) nv = *(const v4f*)(Xn + (size_t)tc * HW + (mb + KTILE) + tm4);
#endif

        // ---- Scores: S(16x16) = Q(16x64) * K(64x16), 16 chained f32 WMMAs (K=4 each)
        v8f S = vzero;
        #pragma unroll
        for (int cb = 0; cb < 16; ++cb) {
            v2f b;  // B[k,m]: lane = m + 16*(k>=2), vgpr = k&1
            b.x = ktile[cur][cb * 4 + coff    ][ml];
            b.y = ktile[cur][cb * 4 + coff + 1][ml];
            S = __builtin_amdgcn_wmma_f32_16x16x4_f32(false, qa[cb], false, b,
                                                      (short)0, S, false, false);
        }

        // ---- Online softmax via DPP row butterflies (VALU, co-executes with WMMA).
        float scale[8];
        #pragma unroll
        for (int v = 0; v < 8; ++v) {
            const float s    = S[v] * LOG2E;            // work in exp2 domain
            const float mnew = fmaxf(mrow[v], rowmax16(s));
            const float sc   = __builtin_amdgcn_exp2f(mrow[v] - mnew);
            const float pv   = __builtin_amdgcn_exp2f(s       - mnew);
            lrow[v]  = lrow[v] * sc + rowsum16(pv);
            mrow[v]  = mnew;
            scale[v] = sc;
            ptile[wave][v + 8 * hi][ml] = pv;           // P[q][m] in D-layout order
        }
        #pragma unroll
        for (int ct = 0; ct < 4; ++ct)
            #pragma unroll
            for (int v = 0; v < 8; ++v)
                acc[ct][v] *= scale[v];

        // ---- PV: out(16x64) += P(16x16) * V(16x64), V[m][c] = ktile[c][m].
        v2f pa[4];  // A of P from LDS (same-wave DS in-order: store->load safe)
        #pragma unroll
        for (int mc = 0; mc < 4; ++mc) {
            pa[mc].x = ptile[wave][ml][mc * 4 + coff    ];
            pa[mc].y = ptile[wave][ml][mc * 4 + coff + 1];
        }
        #pragma unroll
        for (int ct = 0; ct < 4; ++ct) {
            #pragma unroll
            for (int mc = 0; mc < 4; ++mc) {
                v2f b;  // B[m,c]: lane = c + 16*(m>=2), vgpr = m&1
                b.x = ktile[cur][ct * 16 + ml][mc * 4 + coff    ];
                b.y = ktile[cur][ct * 16 + ml][mc * 4 + coff + 1];
                acc[ct] = __builtin_amdgcn_wmma_f32_16x16x4_f32(false, pa[mc], false, b,
                                                                (short)0, acc[ct],
                                                                false, false);
            }
        }

#if HAS_ASYNC_LDS
        // Drain the async fill so the next-buffer data is LDS-visible after the barrier.
        asm volatile("s_wait_asynccnt 0x0" ::: "memory");
#else
        if (pre) *(v4f*)&ktile[cur ^ 1][tc][tm4] = nv;
#endif
        __syncthreads();
    }

    // ---- Epilogue: normalize; each lane owns 8 consecutive q per c -> two b128 stores.
    float rinv[8];
    #pragma unroll
    for (int v = 0; v < 8; ++v) rinv[v] = 1.0f / lrow[v];
    #pragma unroll
    for (int ct = 0; ct < 4; ++ct) {
        const int c = ct * 16 + ml;
        v4f o0, o1;
        o0.x = acc[ct][0] * rinv[0];
        o0.y = acc[ct][1] * rinv[1];
        o0.z = acc[ct][2] * rinv[2];
        o0.w = acc[ct][3] * rinv[3];
        o1.x = acc[ct][4] * rinv[4];
        o1.y = acc[ct][5] * rinv[5];
        o1.z = acc[ct][6] * rinv[6];
        o1.w = acc[ct][7] * rinv[7];
        float* dst = On + (size_t)c * HW + qbase + 8 * hi;
        *(v4f*)(dst    ) = o0;
        *(v4f*)(dst + 4) = o1;
    }
}

extern "C" void kernel_launch(void* const* d_in, const int* in_sizes, int n_in,
                              void* d_out, int out_size, void* d_ws, size_t ws_size,
                              hipStream_t stream) {
    (void)n_in; (void)d_ws; (void)ws_size; (void)out_size;
    const float* X = (const float*)d_in[0];
    float* Out = (float*)d_out;
    const int N = in_sizes[0] / (CDIM * HW);   // = 8
    dim3 grid(HW / (WAVES * QTILE), N, 1);     // (32, 8)
    dim3 block(256, 1, 1);
    RspaA1_attn_kernel<<<grid, block, 0, stream>>>(X, Out);
}